// OnlineTripletLoss_85512798863825
// MI455X (gfx1250) — compile-verified
//
#include <hip/hip_runtime.h>
#include <hip/hip_bf16.h>
#include <math.h>

// OnlineTripletLoss for MI455X (gfx1250, wave32).
// Pipeline: [TDM -> LDS -> WMMA f32 gram] -> [diag+init] -> [dist in-place]
//           -> [triplet accum] -> [finalize]

typedef __attribute__((ext_vector_type(2))) float v2f;
typedef __attribute__((ext_vector_type(8))) float v8f;
typedef unsigned int v4u __attribute__((ext_vector_type(4)));
typedef int          v4i __attribute__((ext_vector_type(4)));
typedef int          v8i __attribute__((ext_vector_type(8)));

#define NPTS 512
#define DIMS 128
#define MARGIN 0.5f
#define EPSPOS 1e-8f

#if defined(__gfx1250__) && __has_builtin(__builtin_amdgcn_tensor_load_to_lds)
#define USE_TDM 1
#endif

// Generic pointer -> LDS byte offset (addrspacecast to AS3, value = LDS offset).
__device__ __forceinline__ unsigned lds_byte_off(const void* p) {
    return (unsigned)(unsigned long long)(__attribute__((address_space(3))) const void*)p;
}

#ifdef USE_TDM
// D# group 0: count=1 | lds_addr[63:32] | global_addr[120:64] | type=2 [127:126]
__device__ __forceinline__ v4u tdm_group0(unsigned ldsaddr, const void* gptr) {
    unsigned long long ga = (unsigned long long)gptr;
    v4u g0;
    g0.x = 1u;                                               // count=1, no gather
    g0.y = ldsaddr;                                          // LDS byte address
    g0.z = (unsigned)(ga & 0xFFFFFFFFull);                   // global_addr[31:0]
    g0.w = (unsigned)((ga >> 32) & 0x01FFFFFFull) | (2u << 30); // [56:32] | type=2
    return g0;
}
// D# group 1: 2D tile, rows x 128 f32, contiguous (stride0 = 128 elements)
__device__ __forceinline__ v8i tdm_group1(int rows) {
    v8i g1;
    g1[0] = (int)(2u << 16);       // workgroup_mask=0, data_size=2 (4 bytes)
    g1[1] = (int)(128u << 16);     // tensor_dim0[15:0]=128 (bits 63:48)
    g1[2] = (int)((unsigned)rows << 16); // tensor_dim1[15:0]=rows (bits 95:80)
    g1[3] = (int)(128u << 16);     // tile_dim0=128 (bits 127:112)
    g1[4] = rows;                  // tile_dim1=rows (bits 143:128), tile_dim2=0
    g1[5] = 128;                   // tensor_dim0_stride[31:0]=128
    g1[6] = 0;                     // stride0[47:32]=0, tensor_dim1_stride lo=0
    g1[7] = 0;
    return g1;
}

__device__ __forceinline__ void tdm_load_2d(unsigned ldsaddr, const void* gptr, int rows) {
    v4u g0 = tdm_group0(ldsaddr, gptr);
    v8i g1 = tdm_group1(rows);
    v4i g2; g2[0] = 1; g2[1] = 1; g2[2] = 0; g2[3] = 0;  // tensor_dim2/3=1, tile_dim3=0
    v4i g3; g3[0] = 0; g3[1] = (int)(1u << 16); g3[2] = 0; g3[3] = 0; // tensor_dim4=1
#if defined(__clang_major__) && (__clang_major__ >= 23)
    v8i g23; g23[0]=g2[0]; g23[1]=g2[1]; g23[2]=g2[2]; g23[3]=g2[3];
    g23[4]=g3[0]; g23[5]=g3[1]; g23[6]=g3[2]; g23[7]=g3[3];
    __builtin_amdgcn_tensor_load_to_lds(g0, g1, g2, g3, g23, 0);
#else
    __builtin_amdgcn_tensor_load_to_lds(g0, g1, g2, g3, 0);
#endif
}
#endif // USE_TDM

// ---------------------------------------------------------------------------
// Kernel 1: G = X * X^T via V_WMMA_F32_16X16X4_F32, tiles staged in LDS by TDM.
// Block = 8 waves; block covers 8 consecutive 16x16 tiles in one tile-row:
//   tm = blockIdx>>2 (shared), tn = (blockIdx*8 & 31) + wave.
// Two TDM descriptors per block: A tile (16 rows, 8 KB) and the contiguous
// 128-row B region (64 KB). Fragments then come from ds_load_b64.
// ---------------------------------------------------------------------------
__global__ __launch_bounds__(256) void gram_wmma_kernel(const float* __restrict__ X,
                                                        float* __restrict__ G) {
    __shared__ float tileA[16 * DIMS];        //  8 KB
    __shared__ float tileB[8 * 16 * DIMS];    // 64 KB

    const int lane = threadIdx.x & 31;
    const int wave = threadIdx.x >> 5;
    const int tb   = (blockIdx.x * 8) & 31;   // first tn tile of this block
    const int tm   = blockIdx.x >> 2;         // tile row (uniform in block)
    const int tn   = tb + wave;               // this wave's tile column
    const int half = lane >> 4;               // 0 or 1
    const int l16  = lane & 15;

#ifdef USE_TDM
    if (wave == 0) {
        tdm_load_2d(lds_byte_off(&tileA[0]), X + (size_t)tm * 16 * DIMS, 16);
        tdm_load_2d(lds_byte_off(&tileB[0]), X + (size_t)tb * 16 * DIMS, 128);
        __builtin_amdgcn_s_wait_tensorcnt(0);
    }
    __syncthreads();
#else
    for (int t = threadIdx.x; t < 16 * DIMS; t += 256)
        tileA[t] = X[(size_t)tm * 16 * DIMS + t];
    for (int t = threadIdx.x; t < 8 * 16 * DIMS; t += 256)
        tileB[t] = X[(size_t)tb * 16 * DIMS + t];
    __syncthreads();
#endif

    // A-fragment (16x4 f32): lanes 0-15 hold K=k,k+1; lanes 16-31 hold K=k+2,k+3.
    // B-fragment of X^T has the identical per-lane pattern on the column tile.
    const float2* As2 = reinterpret_cast<const float2*>(&tileA[l16 * DIMS]);
    const float2* Bs2 = reinterpret_cast<const float2*>(&tileB[(wave * 16 + l16) * DIMS]);

    v8f acc = {};
#pragma unroll
    for (int k = 0; k < DIMS; k += 4) {
        const int h = (k >> 1) + half;        // float2 index of (k + 2*half)
        float2 af = As2[h];
        float2 bf = Bs2[h];
        v2f a; a.x = af.x; a.y = af.y;
        v2f b; b.x = bf.x; b.y = bf.y;
        // (neg_a, A, neg_b, B, c_mod, C, reuse_a, reuse_b)
        acc = __builtin_amdgcn_wmma_f32_16x16x4_f32(false, a, false, b,
                                                    (short)0, acc, false, false);
    }

    // C/D layout: VGPR r -> (M = r + 8*half, N = l16)
#pragma unroll
    for (int r = 0; r < 8; ++r) {
        G[(tm * 16 + r + 8 * half) * NPTS + tn * 16 + l16] = acc[r];
    }
}

// ---------------------------------------------------------------------------
// Kernel 2: sq[i] = G[i,i]; zero the (sum,count) accumulators.
// ---------------------------------------------------------------------------
__global__ void prep_kernel(const float* __restrict__ G,
                            float* __restrict__ sq,
                            float* __restrict__ accum) {
    const int t = threadIdx.x;     // 512 threads
    sq[t] = G[t * (NPTS + 1)];
    if (t < 2) accum[t] = 0.0f;
}

// ---------------------------------------------------------------------------
// Kernel 3: in-place D[i,j] = pre > 0 ? sqrt(pre) : 0, pre = relu(sq_i+sq_j-2G).
// (matches the reference's zero-mask eps trick exactly in value)
// ---------------------------------------------------------------------------
__global__ __launch_bounds__(256) void dist_kernel(float* __restrict__ G,
                                                   const float* __restrict__ sq) {
    const int idx = blockIdx.x * 256 + threadIdx.x;
    const int i = idx >> 9;
    const int j = idx & (NPTS - 1);
    float pre = sq[i] + sq[j] - 2.0f * G[idx];
    pre = fmaxf(pre, 0.0f);
    G[idx] = (pre > 0.0f) ? __builtin_sqrtf(pre) : 0.0f;
}

// ---------------------------------------------------------------------------
// Kernel 4: per-anchor triplet accumulation. One block per anchor i.
// ---------------------------------------------------------------------------
__global__ __launch_bounds__(256) void triplet_kernel(const float* __restrict__ D,
                                                      const int* __restrict__ labels,
                                                      float* __restrict__ accum) {
    __shared__ float drow[NPTS];
    __shared__ int   lab[NPTS];
    __shared__ float ssum[256];
    __shared__ float scnt[256];

    const int i   = blockIdx.x;
    const int tid = threadIdx.x;

    for (int t = tid; t < NPTS; t += 256) {
        drow[t] = D[i * NPTS + t];
        lab[t]  = labels[t];
    }
    __syncthreads();

    const int li = lab[i];
    float s = 0.0f, c = 0.0f;

    for (int j = 0; j < NPTS; ++j) {
        if (j == i || lab[j] != li) continue;          // positives only
        const float dij = drow[j] + MARGIN;
        for (int k = tid; k < NPTS; k += 256) {
            if (lab[k] == li) continue;                // negatives only
            const float v = dij - drow[k];
            const float r = v > 0.0f ? v : 0.0f;
            s += r;
            c += (r > EPSPOS) ? 1.0f : 0.0f;
        }
    }

    ssum[tid] = s;
    scnt[tid] = c;
    __syncthreads();
    for (int off = 128; off > 0; off >>= 1) {
        if (tid < off) {
            ssum[tid] += ssum[tid + off];
            scnt[tid] += scnt[tid + off];
        }
        __syncthreads();
    }
    if (tid == 0) {
        atomicAdd(&accum[0], ssum[0]);
        atomicAdd(&accum[1], scnt[0]);
    }
}

// ---------------------------------------------------------------------------
// Kernel 5: out = sum / num_positives
// ---------------------------------------------------------------------------
__global__ void finalize_kernel(const float* __restrict__ accum, float* __restrict__ out) {
    out[0] = accum[0] / accum[1];
}

extern "C" void kernel_launch(void* const* d_in, const int* in_sizes, int n_in,
                              void* d_out, int out_size, void* d_ws, size_t ws_size,
                              hipStream_t stream) {
    (void)in_sizes; (void)n_in; (void)out_size; (void)ws_size;
    const float* X      = (const float*)d_in[0];   // [512,128] f32
    const int*   labels = (const int*)d_in[1];     // [512] i32
    float*       out    = (float*)d_out;           // scalar f32

    char*  ws    = (char*)d_ws;
    float* G     = (float*)ws;                                 // 1 MB (reused as D)
    float* sq    = (float*)(ws + (size_t)NPTS * NPTS * 4);     // 2 KB
    float* accum = (float*)(ws + (size_t)NPTS * NPTS * 4 + (size_t)NPTS * 4); // 8 B

    gram_wmma_kernel<<<128, 256, 0, stream>>>(X, G);
    prep_kernel<<<1, NPTS, 0, stream>>>(G, sq, accum);
    dist_kernel<<<(NPTS * NPTS) / 256, 256, 0, stream>>>(G, sq);
    triplet_kernel<<<NPTS, 256, 0, stream>>>(G, labels, accum);
    finalize_kernel<<<1, 1, 0, stream>>>(accum, out);
}